// GCN_83983790506410
// MI455X (gfx1250) — compile-verified
//
#include <hip/hip_runtime.h>
#include <hip/hip_bf16.h>
#include <math.h>

// ---------------------------------------------------------------------------
// GCN forward for MI455X (gfx1250, wave32).
// GEMMs use V_WMMA_F32_16X16X4_F32 (full f32 precision, matches reference).
// Graph aggregation via f32 global atomics (L2-resident: all buffers << 192MB).
// ---------------------------------------------------------------------------

typedef float v2f __attribute__((ext_vector_type(2)));
typedef float v8f __attribute__((ext_vector_type(8)));

#define D_IN    128
#define H_DIM   96
#define H2_DIM  48
#define OUT_DIM 16
#define BN_EPS  1e-5f

// ---------------------------------------------------------------------------
// Wave-level f32 WMMA GEMM:  C[M x Nc] = A[M x K] * B[K x Nc]  (row-major)
// One 16x16 C tile per wave, K-loop of 16x16x4 f32 WMMA ops.
// epilogue: 0 = none, 1 = +bias, ReLU   2 = +bias, sigmoid
// Requires: M % 16 == 0, K % 4 == 0, Nc % 16 == 0 (true for all calls here).
// ---------------------------------------------------------------------------
#define GEMM_WAVES 8

__global__ __launch_bounds__(256)
void k_wmma_gemm(const float* __restrict__ A, const float* __restrict__ B,
                 const float* __restrict__ bias, float* __restrict__ C,
                 int M, int K, int Nc, int epilogue) {
    const int lane  = threadIdx.x & 31;
    const int wave  = threadIdx.x >> 5;
    const int rowTile = blockIdx.x * GEMM_WAVES + wave;   // 16-row tile index
    const int colTile = blockIdx.y;                       // 16-col tile index
    if (rowTile * 16 >= M) return;                        // wave-uniform exit

    // ISA 16x4 f32 A layout: lane m=lane&15; lanes<16 hold K={0,1}, lanes>=16 K={2,3}
    const int m     = lane & 15;
    const int khalf = lane >> 4;           // 0 or 1
    const int row   = rowTile * 16 + m;
    const int col   = colTile * 16 + m;    // B/C: N striped across lanes 0..15

    const float* arow = A + (size_t)row * K + 2 * khalf;          // 8B aligned
    const float* bcol = B + (size_t)(2 * khalf) * Nc + col;

    v8f acc = {};
    for (int k = 0; k < K; k += 4) {
        v2f a = *(const v2f*)arow;                 // A[row][k+2*khalf .. +1]
        v2f b;
        b.x = bcol[0];                             // B[k+2*khalf  ][col]
        b.y = bcol[Nc];                            // B[k+2*khalf+1][col]
        acc = __builtin_amdgcn_wmma_f32_16x16x4_f32(
                  /*neg_a=*/false, a, /*neg_b=*/false, b,
                  /*c_mod=*/(short)0, acc, /*reuse_a=*/false, /*reuse_b=*/false);
        arow += 4;
        bcol += (size_t)4 * Nc;
    }

    // C layout: VGPR r -> row (r + 8*khalf), col = lane&15 within tile
    float bv = (epilogue != 0) ? bias[col] : 0.0f;
#pragma unroll
    for (int r = 0; r < 8; ++r) {
        int orow = rowTile * 16 + r + 8 * khalf;
        float v = acc[r];
        if (epilogue == 1) { v += bv; v = fmaxf(v, 0.0f); }
        else if (epilogue == 2) { v += bv; v = 1.0f / (1.0f + __expf(-v)); }
        C[(size_t)orow * Nc + col] = v;
    }
}

// ---------------------------------------------------------------------------
// Graph / BN helper kernels
// ---------------------------------------------------------------------------
__global__ void k_fill(float* __restrict__ p, float v, int n) {
    int i = blockIdx.x * blockDim.x + threadIdx.x;
    if (i < n) p[i] = v;
}

__global__ void k_deg_accum(const int* __restrict__ dst, float* __restrict__ deg, int e) {
    int i = blockIdx.x * blockDim.x + threadIdx.x;
    if (i < e) atomicAdd(&deg[dst[i]], 1.0f);
}

__global__ void k_rsqrt_inplace(float* __restrict__ p, int n) {
    int i = blockIdx.x * blockDim.x + threadIdx.x;
    if (i < n) p[i] = rsqrtf(p[i]);      // deg >= 1 always (self-loop)
}

__global__ void k_edge_norm(const int* __restrict__ src, const int* __restrict__ dst,
                            const float* __restrict__ dinv, float* __restrict__ enorm, int e) {
    int i = blockIdx.x * blockDim.x + threadIdx.x;
    if (i < e) enorm[i] = dinv[src[i]] * dinv[dst[i]];
}

// thread = (edge, 4-feature chunk): gather h[src] float4, atomic-add into agg[dst]
__global__ void k_edge_scatter(const int* __restrict__ src, const int* __restrict__ dst,
                               const float* __restrict__ enorm, const float* __restrict__ h,
                               float* __restrict__ agg, int e) {
    const int chunks = H_DIM / 4;                       // 24
    unsigned tid = blockIdx.x * blockDim.x + threadIdx.x;
    if (tid >= (unsigned)e * chunks) return;
    int ed = tid / chunks;
    int c  = (tid % chunks) * 4;
    float nrm = enorm[ed];
    const float4 hv = *(const float4*)(h + (size_t)src[ed] * H_DIM + c);
    float* ap = agg + (size_t)dst[ed] * H_DIM + c;
    atomicAdd(ap + 0, hv.x * nrm);
    atomicAdd(ap + 1, hv.y * nrm);
    atomicAdd(ap + 2, hv.z * nrm);
    atomicAdd(ap + 3, hv.w * nrm);
}

// agg[i,f] += h[i,f]*dinv[i]^2 + bias[f]; accumulate column sum / sumsq.
// blockDim.x == H_DIM (one thread per column), blocks stride over rows.
__global__ void k_finish_stats(const float* __restrict__ h, const float* __restrict__ dinv,
                               const float* __restrict__ bias, float* __restrict__ agg,
                               float* __restrict__ stats, int n) {
    const int f = threadIdx.x;
    const float bf = bias[f];
    float s = 0.0f, s2 = 0.0f;
    for (int i = blockIdx.x; i < n; i += gridDim.x) {
        float sn = dinv[i]; sn *= sn;
        size_t idx = (size_t)i * H_DIM + f;
        float v = agg[idx] + h[idx] * sn + bf;
        agg[idx] = v;
        s += v; s2 += v * v;
    }
    atomicAdd(&stats[f], s);
    atomicAdd(&stats[H_DIM + f], s2);
}

// out = relu((x - mean) * g * rsqrt(var + eps) + beta), biased var
__global__ void k_bn_relu(const float* __restrict__ agg, const float* __restrict__ stats,
                          const float* __restrict__ g, const float* __restrict__ beta,
                          float* __restrict__ out, int n) {
    unsigned tid = blockIdx.x * blockDim.x + threadIdx.x;
    if (tid >= (unsigned)n * H_DIM) return;
    int f = tid % H_DIM;
    float inv_n = 1.0f / (float)n;
    float mean  = stats[f] * inv_n;
    float var   = stats[H_DIM + f] * inv_n - mean * mean;
    float scale = g[f] * rsqrtf(var + BN_EPS);
    float v = (agg[tid] - mean) * scale + beta[f];
    out[tid] = fmaxf(v, 0.0f);
}

// ---------------------------------------------------------------------------
// Host-side orchestration
// ---------------------------------------------------------------------------
extern "C" void kernel_launch(void* const* d_in, const int* in_sizes, int n_in,
                              void* d_out, int out_size, void* d_ws, size_t ws_size,
                              hipStream_t stream) {
    (void)n_in; (void)out_size; (void)ws_size;

    const int N = in_sizes[0] / D_IN;     // 50000
    const int E = in_sizes[1] / 2;        // 800000

    const float* x   = (const float*)d_in[0];
    const int*   ei  = (const int*)  d_in[1];
    const int*   src = ei;
    const int*   dst = ei + E;
    const float* W1 = (const float*)d_in[2],  *b1 = (const float*)d_in[3];
    const float* g1 = (const float*)d_in[4],  *be1 = (const float*)d_in[5];
    const float* W2 = (const float*)d_in[6],  *b2 = (const float*)d_in[7];
    const float* g2 = (const float*)d_in[8],  *be2 = (const float*)d_in[9];
    const float* W3 = (const float*)d_in[10], *b3 = (const float*)d_in[11];
    const float* g3 = (const float*)d_in[12], *be3 = (const float*)d_in[13];
    const float* fw1 = (const float*)d_in[14], *fb1 = (const float*)d_in[15];
    const float* fw2 = (const float*)d_in[16], *fb2 = (const float*)d_in[17];
    float* out = (float*)d_out;

    // workspace layout (~42 MB)
    float* B0    = (float*)d_ws;                 // N*H
    float* B1    = B0 + (size_t)N * H_DIM;       // N*H
    float* dinv  = B1 + (size_t)N * H_DIM;       // N   (deg, then rsqrt in place)
    float* enorm = dinv + N;                     // E
    float* stats = enorm + E;                    // 2*H

    const int TB = 256;
    const int rowTiles = (N + 15) / 16;                       // 3125
    const int gemmBlocksX = (rowTiles + GEMM_WAVES - 1) / GEMM_WAVES;
    const int nh = N * H_DIM;

    // --- degree normalization (shared by all 3 conv layers) ---
    k_fill<<<(N + TB - 1) / TB, TB, 0, stream>>>(dinv, 1.0f, N);     // self-loop
    k_deg_accum<<<(E + TB - 1) / TB, TB, 0, stream>>>(dst, dinv, E);
    k_rsqrt_inplace<<<(N + TB - 1) / TB, TB, 0, stream>>>(dinv, N);
    k_edge_norm<<<(E + TB - 1) / TB, TB, 0, stream>>>(src, dst, dinv, enorm, E);

    // --- one GCN layer: gemm -> scatter -> finish(+bias,+self,stats) -> bn_relu ---
    auto layer = [&](const float* act_in, int K, const float* W, const float* b,
                     const float* g, const float* be, float* G, float* Ag) {
        dim3 gg(gemmBlocksX, H_DIM / 16);
        k_wmma_gemm<<<gg, TB, 0, stream>>>(act_in, W, nullptr, G, N, K, H_DIM, 0);
        k_fill<<<(nh + TB - 1) / TB, TB, 0, stream>>>(Ag, 0.0f, nh);
        k_fill<<<1, TB, 0, stream>>>(stats, 0.0f, 2 * H_DIM);
        int nscatter = E * (H_DIM / 4);
        k_edge_scatter<<<(nscatter + TB - 1) / TB, TB, 0, stream>>>(src, dst, enorm, G, Ag, E);
        k_finish_stats<<<512, H_DIM, 0, stream>>>(G, dinv, b, Ag, stats, N);
        k_bn_relu<<<(nh + TB - 1) / TB, TB, 0, stream>>>(Ag, stats, g, be, G, N);
        // activation now lives in G
    };

    layer(x,  D_IN,  W1, b1, g1, be1, B0, B1);   // act1 -> B0
    layer(B0, H_DIM, W2, b2, g2, be2, B1, B0);   // act2 -> B1
    layer(B1, H_DIM, W3, b3, g3, be3, B0, B1);   // act3 -> B0

    // --- MLP head (fused bias+activation epilogues) ---
    dim3 gh1(gemmBlocksX, H2_DIM / 16);
    k_wmma_gemm<<<gh1, TB, 0, stream>>>(B0, fw1, fb1, B1, N, H_DIM, H2_DIM, 1);  // relu
    dim3 gh2(gemmBlocksX, OUT_DIM / 16);
    k_wmma_gemm<<<gh2, TB, 0, stream>>>(B1, fw2, fb2, out, N, H2_DIM, OUT_DIM, 2); // sigmoid
}